// MoE_68719477270
// MI455X (gfx1250) — compile-verified
//
#include <hip/hip_runtime.h>
#include <hip/hip_bf16.h>

typedef __attribute__((ext_vector_type(2))) float v2f;
typedef __attribute__((ext_vector_type(8))) float v8f;
typedef __attribute__((ext_vector_type(4))) int   v4i;

#define N_TOK 8192
#define DIM   2048
#define NEXP  8

#define TM 64           // tokens per tile
#define TN 64           // output cols per tile
#define KC 32           // K-chunk staged in LDS
#define ASP 36          // A LDS row stride (floats): 16B aligned, conflict-free
#define KCP 34          // B (transposed) LDS row stride (floats): 8B aligned

// ---- CDNA5 async global->LDS path (guarded; falls back to sync) ----
#if defined(__has_builtin)
#  if __has_builtin(__builtin_amdgcn_global_load_async_to_lds_b128) && \
      __has_builtin(__builtin_amdgcn_global_load_async_to_lds_b32)
#    define HAVE_ASYNC_LDS 1
#  endif
#endif
#ifndef HAVE_ASYNC_LDS
#  define HAVE_ASYNC_LDS 0
#endif

#if HAVE_ASYNC_LDS
#  define GAS __attribute__((address_space(1)))
#  define LAS __attribute__((address_space(3)))
#  define G128(p) ((GAS v4i*)(void*)(p))
#  define L128(p) ((LAS v4i*)(void*)(p))
#  define G32(p)  ((GAS int*)(void*)(p))
#  define L32(p)  ((LAS int*)(void*)(p))
#  if __has_builtin(__builtin_amdgcn_s_wait_asynccnt)
#    define WAIT_ASYNC(n) __builtin_amdgcn_s_wait_asynccnt(n)
#  else
#    define WAIT_ASYNC(n) asm volatile("s_wait_asynccnt %0" :: "i"(n))
#  endif
#  define NASYNC 10   // async instructions issued per wave per chunk (2 A + 8 B)
#endif

// ---------------------------------------------------------------------------
// Gating: one wave32 per token. logits = x . Wg[:,e], top-2, softmax over the
// two selected logits, atomic compaction into per-expert token/weight lists.
// ---------------------------------------------------------------------------
__launch_bounds__(256)
__global__ void gate_topk(const float* __restrict__ X,
                          const float* __restrict__ Wg,
                          int* __restrict__ counts,
                          int* __restrict__ idx,
                          float* __restrict__ wgt)
{
    const int gwave = (blockIdx.x * blockDim.x + threadIdx.x) >> 5;
    const int lane  = threadIdx.x & 31;
    if (gwave >= N_TOK) return;

    const float* x = X + (size_t)gwave * DIM;

    float acc[NEXP];
#pragma unroll
    for (int e = 0; e < NEXP; ++e) acc[e] = 0.f;

    for (int d = lane; d < DIM; d += 32) {
        const float xv = x[d];
        const float4* wr = (const float4*)(Wg + (size_t)d * NEXP);
        const float4 w0 = wr[0];
        const float4 w1 = wr[1];
        acc[0] += xv * w0.x; acc[1] += xv * w0.y;
        acc[2] += xv * w0.z; acc[3] += xv * w0.w;
        acc[4] += xv * w1.x; acc[5] += xv * w1.y;
        acc[6] += xv * w1.z; acc[7] += xv * w1.w;
    }

#pragma unroll
    for (int off = 16; off > 0; off >>= 1) {
#pragma unroll
        for (int e = 0; e < NEXP; ++e)
            acc[e] += __shfl_xor(acc[e], off, 32);
    }

    if (lane == 0) {
        int e1 = 0; float m1 = acc[0];
#pragma unroll
        for (int e = 1; e < NEXP; ++e)
            if (acc[e] > m1) { m1 = acc[e]; e1 = e; }
        int e2 = -1; float m2 = -3.4e38f;
#pragma unroll
        for (int e = 0; e < NEXP; ++e)
            if (e != e1 && acc[e] > m2) { m2 = acc[e]; e2 = e; }

        const float ex  = __expf(m2 - m1);
        const float den = 1.f / (1.f + ex);

        int p1 = atomicAdd(&counts[e1], 1);
        idx[e1 * N_TOK + p1] = gwave;
        wgt[e1 * N_TOK + p1] = den;
        int p2 = atomicAdd(&counts[e2], 1);
        idx[e2 * N_TOK + p2] = gwave;
        wgt[e2 * N_TOK + p2] = ex * den;
    }
}

// ---------------------------------------------------------------------------
// Grouped GEMM over routed token lists with fp32 WMMA 16x16x4.
// A staged [m][k]; B staged TRANSPOSED [n][k] so both fragments are
// contiguous 8B LDS loads. Async double-buffered staging when available.
// ---------------------------------------------------------------------------
__launch_bounds__(256, 2)
__global__ void moe_gemm(const float* __restrict__ X,
                         const float* __restrict__ We,
                         const float* __restrict__ be,
                         const int*   __restrict__ counts,
                         const int*   __restrict__ idx,
                         const float* __restrict__ wgt,
                         float* __restrict__ out)
{
    const int e  = blockIdx.z;
    const int m0 = blockIdx.y * TM;
    const int n0 = blockIdx.x * TN;
    const int cnt = counts[e];
    if (m0 >= cnt) return;                 // block-uniform exit (EXEC stays full)

    __shared__ float As[2][TM][ASP];       // activations [m][k]
    __shared__ float Bs[2][TN][KCP];       // weights transposed [n][k]
    __shared__ int   rid[TM];
    __shared__ float rw[TM];

    const int tid = threadIdx.x;
    if (tid < TM) {
        const int r = m0 + tid;
        if (r < cnt) {
            rid[tid] = idx[e * N_TOK + r];
            rw[tid]  = wgt[e * N_TOK + r];
        } else {
            rid[tid] = 0;
            rw[tid]  = 0.f;                // padded rows contribute exact zero
        }
    }
    __syncthreads();

    const int lane = tid & 31;
    const int wave = tid >> 5;
    const int wm   = wave & 3;             // 4 m-subtiles of 16
    const int wn   = wave >> 2;            // 2 n-halves of 32
    const int lm   = lane & 15;
    const int kh   = (lane >> 4) << 1;     // K offset {0,2}: ISA f32 A/B lane layout

    // staging coordinates
    const int ar = tid >> 2, ac = (tid & 3) * 8;     // A: 4 thr/row, 8 floats
    const size_t arow = (size_t)rid[ar] * DIM;       // gathered token row
    const int bn = tid & 63, bk = tid >> 6;          // async B: 1 float/lane, transposing
    const int br = tid >> 3, bc = (tid & 7) * 8;     // sync  B: 8 floats/thr

    const float* Wexp = We + (size_t)e * DIM * DIM;

    v8f acc0 = {};                         // cols n0 + wn*32 + [0,16)
    v8f acc1 = {};                         // cols n0 + wn*32 + [16,32)

    auto compute = [&](int buf) {
#pragma unroll
        for (int kk = 0; kk < KC; kk += 4) {
            v2f a  = *(const v2f*)&As[buf][wm * 16 + lm][kk + kh];
            v2f b0 = *(const v2f*)&Bs[buf][wn * 32 + lm][kk + kh];
            v2f b1 = *(const v2f*)&Bs[buf][wn * 32 + 16 + lm][kk + kh];
            acc0 = __builtin_amdgcn_wmma_f32_16x16x4_f32(
                       false, a, false, b0, (short)0, acc0, false, false);
            acc1 = __builtin_amdgcn_wmma_f32_16x16x4_f32(
                       false, a, false, b1, (short)0, acc1, false, false);
        }
    };

#if HAVE_ASYNC_LDS
    // -------- async double-buffered pipeline (ASYNCcnt-tracked) --------
    auto stage = [&](int buf, int k0) {
        const float* ga = X + arow + k0 + ac;
        __builtin_amdgcn_global_load_async_to_lds_b128(
            G128(ga),     L128(&As[buf][ar][ac]),     0, 0);
        __builtin_amdgcn_global_load_async_to_lds_b128(
            G128(ga + 4), L128(&As[buf][ar][ac + 4]), 0, 0);
#pragma unroll
        for (int r = 0; r < 8; ++r) {      // transpose on the fly: [k][n] -> [n][k]
            const int k = bk + r * 4;
            __builtin_amdgcn_global_load_async_to_lds_b32(
                G32(Wexp + (size_t)(k0 + k) * DIM + n0 + bn),
                L32(&Bs[buf][bn][k]), 0, 0);
        }
    };

    stage(0, 0);
    for (int k0 = 0, buf = 0; k0 < DIM; k0 += KC, buf ^= 1) {
        if (k0 + KC < DIM) {
            stage(buf ^ 1, k0 + KC);       // prefetch next chunk into other buffer
            WAIT_ASYNC(NASYNC);            // drain current chunk, keep next in flight
        } else {
            WAIT_ASYNC(0);
        }
        __syncthreads();
        compute(buf);
        __syncthreads();                   // buffer reuse fence before next issue
    }
#else
    // -------- synchronous fallback --------
    for (int k0 = 0; k0 < DIM; k0 += KC) {
        __syncthreads();
        {
            const float* ga = X + arow + k0 + ac;
            const float4 a0 = ((const float4*)ga)[0];
            const float4 a1 = ((const float4*)ga)[1];
            *(float4*)&As[0][ar][ac]     = a0;
            *(float4*)&As[0][ar][ac + 4] = a1;
        }
        {
            const float* gb = Wexp + (size_t)(k0 + br) * DIM + n0 + bc;
            const float4 b0 = ((const float4*)gb)[0];
            const float4 b1 = ((const float4*)gb)[1];
            Bs[0][bc + 0][br] = b0.x; Bs[0][bc + 1][br] = b0.y;
            Bs[0][bc + 2][br] = b0.z; Bs[0][bc + 3][br] = b0.w;
            Bs[0][bc + 4][br] = b1.x; Bs[0][bc + 5][br] = b1.y;
            Bs[0][bc + 6][br] = b1.z; Bs[0][bc + 7][br] = b1.w;
        }
        __syncthreads();
        compute(0);
    }
#endif

    // Epilogue: out[token] += coeff * (acc + bias). C/D: lanes 0-15 hold
    // N=lane, M=v; lanes 16-31 hold N=lane-16, M=v+8.
    const int mbase = wm * 16 + ((lane >> 4) << 3);
    const int col0  = n0 + wn * 32 + lm;
    const int col1  = col0 + 16;
    const float bias0 = be[(size_t)e * DIM + col0];
    const float bias1 = be[(size_t)e * DIM + col1];

#pragma unroll
    for (int v = 0; v < 8; ++v) {
        const int   mr  = mbase + v;
        const int   tok = rid[mr];
        const float w   = rw[mr];
        atomicAdd(&out[(size_t)tok * DIM + col0], w * (acc0[v] + bias0));
        atomicAdd(&out[(size_t)tok * DIM + col1], w * (acc1[v] + bias1));
    }
}

// ---------------------------------------------------------------------------
extern "C" void kernel_launch(void* const* d_in, const int* in_sizes, int n_in,
                              void* d_out, int out_size, void* d_ws, size_t ws_size,
                              hipStream_t stream)
{
    const float* X  = (const float*)d_in[0];  // [N, D]
    const float* Wg = (const float*)d_in[1];  // [D, E]
    const float* We = (const float*)d_in[2];  // [E, D, D]
    const float* be = (const float*)d_in[3];  // [E, D]
    float* out = (float*)d_out;               // [N, D]

    int*   counts = (int*)d_ws;
    int*   idx    = (int*)((char*)d_ws + 256);
    float* wgt    = (float*)((char*)d_ws + 256 + (size_t)NEXP * N_TOK * sizeof(int));

    (void)hipMemsetAsync(counts, 0, 256, stream);
    (void)hipMemsetAsync(d_out, 0, (size_t)out_size * sizeof(float), stream);

    gate_topk<<<N_TOK / 8, 256, 0, stream>>>(X, Wg, counts, idx, wgt);

    dim3 grid(DIM / TN, N_TOK / TM, NEXP);
    moe_gemm<<<grid, 256, 0, stream>>>(X, We, be, counts, idx, wgt, out);
}